// ALE_40286793237130
// MI455X (gfx1250) — compile-verified
//
#include <hip/hip_runtime.h>

// ext-vector types for WMMA operands
typedef __attribute__((ext_vector_type(2))) float v2f;
typedef __attribute__((ext_vector_type(8))) float v8f;

#define BATCH   512
#define IMG_D   1024
#define WORD_D  512
#define NCLS    20000

// ---------------------------------------------------------------------------
// CDNA5 async global->LDS copy (ASYNCcnt-tracked), probed via __has_builtin so
// the file always compiles; falls back to plain load+ds_store otherwise.
// Probe-confirmed signature (from hipcc diagnostic):
//   void __builtin_amdgcn_global_load_async_to_lds_b128(
//       v4i addrspace(1)* gsrc, v4i addrspace(3)* ldst, imm int off, imm int cpol)
// where v4i = int __attribute__((vector_size(16))).
// ---------------------------------------------------------------------------
#if defined(__AMDGCN__) && \
    __has_builtin(__builtin_amdgcn_global_load_async_to_lds_b128) && \
    __has_builtin(__builtin_amdgcn_s_wait_asynccnt)
#define USE_ASYNC_LDS 1
typedef int v4i __attribute__((vector_size(4 * sizeof(int))));
typedef __attribute__((address_space(1))) v4i g_v4i;
typedef __attribute__((address_space(3))) v4i l_v4i;
#else
#define USE_ASYNC_LDS 0
#endif

__device__ __forceinline__ void copy16_to_lds(const float* gsrc, float* ldst) {
#if USE_ASYNC_LDS
    __builtin_amdgcn_global_load_async_to_lds_b128(
        (g_v4i*)(const void*)gsrc, (l_v4i*)(void*)ldst, 0, 0);
#else
    *(float4*)ldst = *(const float4*)gsrc;
#endif
}

__device__ __forceinline__ void wait_async_copies() {
#if USE_ASYNC_LDS
    __builtin_amdgcn_s_wait_asynccnt(0);
#endif
}

// ---------------------------------------------------------------------------
// GEMM1: xW[512x512] = x[512x1024] @ W[1024x512]   (fp32, V_WMMA_F32_16X16X4_F32)
// Block: 256 thr (8 waves). Tile 64(M) x 64(N), K-chunk 32. (0.5 GFLOP; simple.)
// ---------------------------------------------------------------------------
__global__ __launch_bounds__(256)
void ALE_gemm1_xw(const float* __restrict__ X,   // [512 x 1024]
                  const float* __restrict__ W,   // [1024 x 512]
                  float* __restrict__ C) {       // [512 x 512]
    __shared__ float As[64 * 36];   // m-major, padded stride 36
    __shared__ float Bs[32 * 68];   // k-major, padded stride 68

    const int tid  = threadIdx.x;
    const int wave = tid >> 5;
    const int lane = tid & 31;
    const int m    = lane & 15;
    const int hi   = lane >> 4;
    const int wmb  = (wave & 3) * 16;
    const int wnb  = (wave >> 2) * 32;
    const int m0   = blockIdx.x * 64;
    const int n0   = blockIdx.y * 64;

    v8f acc0 = {};
    v8f acc1 = {};

    const int arow = tid >> 3;          // 0..31
    const int ac4  = (tid & 7) * 4;     // 0..28
    const int brow = tid >> 4;          // 0..15
    const int bc4  = (tid & 15) * 4;    // 0..60

    for (int kc = 0; kc < IMG_D; kc += 32) {
#pragma unroll
        for (int it = 0; it < 2; ++it) {
            int r = arow + it * 32;
            copy16_to_lds(&X[(m0 + r) * IMG_D + kc + ac4], &As[r * 36 + ac4]);
        }
#pragma unroll
        for (int it = 0; it < 2; ++it) {
            int r = brow + it * 16;
            copy16_to_lds(&W[(kc + r) * WORD_D + n0 + bc4], &Bs[r * 68 + bc4]);
        }
        wait_async_copies();
        __syncthreads();

#pragma unroll
        for (int kk = 0; kk < 32; kk += 4) {
            const int ka = kk + 2 * hi;
            v2f a = *(const v2f*)&As[(wmb + m) * 36 + ka];
            {
                int col = wnb + m;
                v2f b;
                b.x = Bs[ka * 68 + col];
                b.y = Bs[(ka + 1) * 68 + col];
                acc0 = __builtin_amdgcn_wmma_f32_16x16x4_f32(
                    false, a, false, b, (short)0, acc0, false, false);
            }
            {
                int col = wnb + 16 + m;
                v2f b;
                b.x = Bs[ka * 68 + col];
                b.y = Bs[(ka + 1) * 68 + col];
                acc1 = __builtin_amdgcn_wmma_f32_16x16x4_f32(
                    false, a, false, b, (short)0, acc1, false, false);
            }
        }
        __syncthreads();
    }

    const int rbase = m0 + wmb + 8 * hi;
#pragma unroll
    for (int r = 0; r < 8; ++r) {
        C[(rbase + r) * WORD_D + n0 + wnb + m]      = acc0[r];
        C[(rbase + r) * WORD_D + n0 + wnb + 16 + m] = acc1[r];
    }
}

// ---------------------------------------------------------------------------
// GEMM2: scores[512x20000] = xW[512x512] @ E^T   (dominant GEMM, 10.5 GFLOP)
// Block: 256 thr (8 waves). Tile 128(M) x 160(N), K-chunk 32, DOUBLE-BUFFERED
// LDS with async global->LDS prefetch of chunk k+1 overlapped with WMMA on k.
// Grid: (512/128) x (20000/160) = 4 x 125 — exact, no guards.
// ---------------------------------------------------------------------------
__global__ __launch_bounds__(256)
void ALE_gemm2_scores(const float* __restrict__ A,   // xW [512 x 512]
                      const float* __restrict__ E,   // [20000 x 512]
                      float* __restrict__ C) {       // [512 x 20000]
    __shared__ float As[2][128 * 36];   // m-major, padded stride 36
    __shared__ float Bs[2][160 * 36];   // n-major (E rows), padded stride 36

    const int tid  = threadIdx.x;
    const int wave = tid >> 5;
    const int lane = tid & 31;
    const int m    = lane & 15;
    const int hi   = lane >> 4;
    const int wmb  = (wave & 3) * 32;   // 0,32,64,96
    const int wnb  = (wave >> 2) * 80;  // 0,80
    const int m0   = blockIdx.x * 128;
    const int n0   = blockIdx.y * 160;

    v8f acc[2][5];
    {
        v8f z = {};
#pragma unroll
        for (int i = 0; i < 2; ++i)
#pragma unroll
            for (int j = 0; j < 5; ++j) acc[i][j] = z;
    }

    const int arow = tid >> 3;          // 0..31
    const int ac4  = (tid & 7) * 4;     // 0..28

    // tile loader: A 128x32 (4 x b128/thread), E 160x32 (5 x b128/thread)
    auto load_tiles = [&](int kc, int buf) {
#pragma unroll
        for (int it = 0; it < 4; ++it) {
            int r = arow + it * 32;
            copy16_to_lds(&A[(m0 + r) * WORD_D + kc + ac4],
                          &As[buf][r * 36 + ac4]);
        }
#pragma unroll
        for (int it = 0; it < 5; ++it) {
            int r = arow + it * 32;
            copy16_to_lds(&E[(size_t)(n0 + r) * WORD_D + kc + ac4],
                          &Bs[buf][r * 36 + ac4]);
        }
    };

    // prologue: fill buffer 0
    load_tiles(0, 0);
    wait_async_copies();
    __syncthreads();

    int cur = 0;
    for (int kc = 0; kc < WORD_D; kc += 32) {
        const int nxt = cur ^ 1;
        // issue async prefetch of next chunk while we compute on `cur`
        if (kc + 32 < WORD_D) load_tiles(kc + 32, nxt);

        const float* Ac = As[cur];
        const float* Bc = Bs[cur];
#pragma unroll
        for (int kk = 0; kk < 32; kk += 4) {
            const int ka = kk + 2 * hi;
            v2f a[2], b[5];
#pragma unroll
            for (int i = 0; i < 2; ++i)
                a[i] = *(const v2f*)&Ac[(wmb + i * 16 + m) * 36 + ka];
#pragma unroll
            for (int j = 0; j < 5; ++j)
                b[j] = *(const v2f*)&Bc[(wnb + j * 16 + m) * 36 + ka];
#pragma unroll
            for (int i = 0; i < 2; ++i)
#pragma unroll
                for (int j = 0; j < 5; ++j)
                    acc[i][j] = __builtin_amdgcn_wmma_f32_16x16x4_f32(
                        false, a[i], false, b[j], (short)0, acc[i][j],
                        false, false);
        }

        wait_async_copies();   // next buffer fully landed in LDS
        __syncthreads();       // all waves done reading `cur` / see `nxt`
        cur = nxt;
    }

    // Store accumulators
#pragma unroll
    for (int i = 0; i < 2; ++i) {
        const int rbase = m0 + wmb + i * 16 + 8 * hi;
#pragma unroll
        for (int j = 0; j < 5; ++j) {
            const int col = n0 + wnb + j * 16 + m;
#pragma unroll
            for (int r = 0; r < 8; ++r)
                C[(size_t)(rbase + r) * NCLS + col] = acc[i][j][r];
        }
    }
}

// ---------------------------------------------------------------------------
extern "C" void kernel_launch(void* const* d_in, const int* in_sizes, int n_in,
                              void* d_out, int out_size, void* d_ws, size_t ws_size,
                              hipStream_t stream) {
    const float* x = (const float*)d_in[0];   // [512 x 1024]
    const float* E = (const float*)d_in[1];   // [20000 x 512]
    const float* W = (const float*)d_in[2];   // [1024 x 512]
    float* out = (float*)d_out;               // [512 x 20000]
    float* xW  = (float*)d_ws;                // [512 x 512] scratch (1 MB)

    ALE_gemm1_xw<<<dim3(8, 8), 256, 0, stream>>>(x, W, xW);
    ALE_gemm2_scores<<<dim3(4, 125), 256, 0, stream>>>(xW, E, out);
}